// TimeNetDecoder_31679678776047
// MI455X (gfx1250) — compile-verified
//
#include <hip/hip_runtime.h>

// ---------------------------------------------------------------------------
// GRU decoder for MI455X (gfx1250):
//  - fp32 WMMA (v_wmma_f32_16x16x4_f32) for recurrent GEMMs + projection head
//  - Tensor Data Mover (tensor_load_to_lds) double-buffered A-tile staging,
//    using TDM hardware padding to produce a bank-conflict-free skewed layout
//  - weights stay L2-resident across all 512 steps; per-step HBM traffic is
//    just the 0.4MB h-state plus the final masked output slice
// ---------------------------------------------------------------------------

typedef float v2f __attribute__((ext_vector_type(2)));
typedef float v8f __attribute__((ext_vector_type(8)));
typedef unsigned int u32x4 __attribute__((ext_vector_type(4)));
typedef int i32x4 __attribute__((ext_vector_type(4)));
typedef int i32x8 __attribute__((ext_vector_type(8)));

#define WMMA_F32(a, b, c) \
  __builtin_amdgcn_wmma_f32_16x16x4_f32(false, (a), false, (b), (short)0, (c), false, false)

#if defined(__AMDGCN__) && __has_builtin(__builtin_amdgcn_tensor_load_to_lds) && \
    __has_builtin(__builtin_amdgcn_s_wait_tensorcnt)
#define USE_TDM 1
#else
#define USE_TDM 0
#endif

constexpr int kB = 128;    // batch
constexpr int kH = 768;    // hidden
constexpr int kF = 128;    // output features
constexpr int kS = 512;    // MAX_SEQ_LEN
constexpr int kSE = 32;    // S_ENC
constexpr int kG3 = 3 * kH;   // 2304 (one gate block)
constexpr int kGW = 6 * kH;   // 4608 (gi | gh row width)

constexpr int kKC = 128;              // K-chunk (power of two: TDM pad interval)
constexpr int kLdsStride = kKC + 4;   // 132: 132%64==4 -> 16 lanes hit distinct bank pairs
constexpr int kChunkFloats = 32 * kLdsStride;          // 4224 floats per buffer
constexpr unsigned kGatesLdsBytes = 2u * kChunkFloats * 4u;  // 33792 B double buffer

// --- copy x0 = H_o[:, -1, :] into contiguous scratch --------------------------
__global__ void init_x0(const float* __restrict__ H_o, float* __restrict__ x0) {
  int idx = blockIdx.x * blockDim.x + threadIdx.x;
  if (idx < kB * kH) {
    int b = idx / kH, j = idx % kH;
    x0[idx] = H_o[((size_t)b * kSE + (kSE - 1)) * kH + j];
  }
}

// --- per-step gates GEMM: G[128 x 4608] = [x@W_ih^T + b_ih | h@W_hh^T + b_hh] -
// Grid: 288 blocks = 4 row-groups (32 rows) x 72 col-groups (64 cols).
// Block: 128 threads = 4 waves; each wave owns one 16-col tile x 2 row tiles.
__global__ __launch_bounds__(128) void gru_gates(
    const float* __restrict__ x, const float* __restrict__ h,
    const float* __restrict__ W_ih, const float* __restrict__ W_hh,
    const float* __restrict__ b_ih, const float* __restrict__ b_hh,
    float* __restrict__ Gmat) {
  extern __shared__ float ldsA[];  // 2 buffers of 32 x 132 floats

  const int tid  = threadIdx.x;
  const int wave = tid >> 5;
  const int lane = tid & 31;
  const int rowGroup = blockIdx.x & 3;   // 0..3  -> rows rowGroup*32
  const int colGroup = blockIdx.x >> 2;  // 0..71
  const int side = (colGroup >= 36) ? 1 : 0;  // 0: gi (x,W_ih)  1: gh (h,W_hh)

  const float* __restrict__ A    = side ? h    : x;
  const float* __restrict__ W    = side ? W_hh : W_ih;
  const float* __restrict__ bias = side ? b_hh : b_ih;
  const int colW    = (colGroup - side * 36) * 64 + wave * 16;  // tile col in W-space
  const int rowBase = rowGroup * 32;

  const int m  = lane & 15;          // row (A) / col (B) within tile
  const int ko = (lane >> 4) * 2;    // k sub-offset: lanes 16..31 hold k+2,k+3

  v8f acc0 = {};  // rows rowBase + 0..15
  v8f acc1 = {};  // rows rowBase + 16..31

#if USE_TDM
  // One TDM descriptor moves the 32x128 activation tile (row stride 768) into
  // LDS; pad_interval=128 DW + pad_amount=4 DW => skewed row stride of 132.
  auto issue_tdm = [&](int kc, int buf) {
    if (wave == 0) {
      unsigned long long gaddr =
          (unsigned long long)(const void*)(A + (size_t)rowBase * kH + kc);
      unsigned int laddr =
          (unsigned int)(unsigned long long)(const void*)(ldsA + buf * kChunkFloats);
      u32x4 g0;
      g0[0] = 1u;                                   // count=1 valid descriptor
      g0[1] = laddr;                                // lds_addr
      g0[2] = (unsigned int)gaddr;                  // global_addr[31:0]
      g0[3] = (unsigned int)((gaddr >> 32) & 0x01FFFFFFu) | (2u << 30);  // [56:32] | type=2
      i32x8 g1;
      g1[0] = (2 << 16)      // data_size = 4B
            | (1 << 20)      // pad_enable
            | (6 << 22)      // pad_interval: 128 DWORDs
            | (3 << 25);     // pad_amount:   4 DWORDs
      g1[1] = (int)(((unsigned)kH & 0xFFFFu) << 16);   // tensor_dim0 lo16 (768)
      g1[2] = (int)(((unsigned)kB & 0xFFFFu) << 16);   // dim0 hi=0 | tensor_dim1 lo16 (128)
      g1[3] = (int)(((unsigned)kKC) << 16);            // dim1 hi=0 | tile_dim0 (128)
      g1[4] = 32;                                      // tile_dim1=32, tile_dim2=0
      g1[5] = kH;                                      // tensor_dim0_stride lo32 (768)
      g1[6] = 0;
      g1[7] = 0;
      i32x4 z4 = {0, 0, 0, 0};                         // groups 2/3: NULL (<=2D tensor)
      i32x8 z8 = {0, 0, 0, 0, 0, 0, 0, 0};
      __builtin_amdgcn_tensor_load_to_lds(g0, g1, z4, z4, z8, 0);
    }
  };

  issue_tdm(0, 0);
  if (wave == 0) __builtin_amdgcn_s_wait_tensorcnt(0);
  __syncthreads();

  int buf = 0;
  for (int kc = 0; kc < kH; kc += kKC) {
    const bool more = (kc + kKC) < kH;
    if (more) issue_tdm(kc + kKC, buf ^ 1);   // overlap TDM with WMMA compute

    const float* __restrict__ As = ldsA + buf * kChunkFloats;
    const float* __restrict__ Wrow = W + (size_t)(colW + m) * kH + kc;
#pragma unroll 8
    for (int k0 = 0; k0 < kKC; k0 += 4) {
      const int kk = k0 + ko;                                // even -> 8B aligned
      v2f a0 = *(const v2f*)&As[m * kLdsStride + kk];        // ds_load_b64
      v2f a1 = *(const v2f*)&As[(m + 16) * kLdsStride + kk];
      v2f bb = *(const v2f*)&Wrow[kk];                       // global_load_b64 (L2)
      acc0 = WMMA_F32(a0, bb, acc0);
      acc1 = WMMA_F32(a1, bb, acc1);
    }

    if (more) {
      if (wave == 0) __builtin_amdgcn_s_wait_tensorcnt(0);
      __syncthreads();
    }
    buf ^= 1;
  }
#else
  // Fallback: synchronous cooperative staging into the same skewed layout.
  for (int kc = 0; kc < kH; kc += kKC) {
    for (int i = tid; i < 32 * kKC; i += 128) {
      int r = i >> 7;        // /128
      int c = i & (kKC - 1);
      ldsA[r * kLdsStride + c] = A[(size_t)(rowBase + r) * kH + kc + c];
    }
    __syncthreads();
    const float* __restrict__ Wrow = W + (size_t)(colW + m) * kH + kc;
#pragma unroll 8
    for (int k0 = 0; k0 < kKC; k0 += 4) {
      const int kk = k0 + ko;
      v2f a0 = *(const v2f*)&ldsA[m * kLdsStride + kk];
      v2f a1 = *(const v2f*)&ldsA[(m + 16) * kLdsStride + kk];
      v2f bb = *(const v2f*)&Wrow[kk];
      acc0 = WMMA_F32(a0, bb, acc0);
      acc1 = WMMA_F32(a1, bb, acc1);
    }
    __syncthreads();
  }
#endif

  // D layout: lane l, reg r -> row = r + 8*(l>>4), col = l&15
  const float bv   = bias[colW + m];
  const int   gcol = side * kG3 + colW + m;
  const int   rhi  = (lane >> 4) * 8;
#pragma unroll
  for (int r = 0; r < 8; ++r) {
    int row0 = rowBase + rhi + r;
    Gmat[(size_t)row0 * kGW + gcol]        = acc0[r] + bv;
    Gmat[(size_t)(row0 + 16) * kGW + gcol] = acc1[r] + bv;
  }
}

// --- per-step combine + projection + reverse + mask ---------------------------
// Grid: 8 blocks (16 batch rows each). Block: 256 threads = 8 waves.
__global__ __launch_bounds__(256) void gru_combine_project(
    const float* __restrict__ Gmat, const float* __restrict__ hPrev,
    float* __restrict__ hOut,
    const float* __restrict__ W_lin, const float* __restrict__ b_lin,
    const int* __restrict__ T, float* __restrict__ out, int s) {
  __shared__ float ldsH[16 * 772];  // padded stride 772 (4-bank skew)

  const int tid     = threadIdx.x;
  const int rowBase = blockIdx.x * 16;

  // ---- phase 1: elementwise gate combine; write h_new to LDS + global ----
  for (int i = tid; i < 16 * kH; i += 256) {
    int r = i / kH, j = i % kH;
    int b = rowBase + r;
    const float* Grow = Gmat + (size_t)b * kGW;
    float gi_r = Grow[j];
    float gi_z = Grow[kH + j];
    float gi_n = Grow[2 * kH + j];
    float gh_r = Grow[kG3 + j];
    float gh_z = Grow[kG3 + kH + j];
    float gh_n = Grow[kG3 + 2 * kH + j];
    float rg = 1.0f / (1.0f + __expf(-(gi_r + gh_r)));
    float zg = 1.0f / (1.0f + __expf(-(gi_z + gh_z)));
    float ng = tanhf(gi_n + rg * gh_n);
    float hp = hPrev[(size_t)b * kH + j];
    float hn = (1.0f - zg) * ng + zg * hp;
    ldsH[r * 772 + j]        = hn;
    hOut[(size_t)b * kH + j] = hn;
  }
  __syncthreads();

  // ---- phase 2: 16x128 projection tile; 8 waves -> 8 col tiles of 16 ----
  const int wave = tid >> 5;
  const int lane = tid & 31;
  const int m    = lane & 15;
  const int ko   = (lane >> 4) * 2;
  const int fcol = wave * 16;

  v8f acc = {};
  const float* __restrict__ Wrow = W_lin + (size_t)(fcol + m) * kH;
#pragma unroll 8
  for (int k0 = 0; k0 < kH; k0 += 4) {
    const int kk = k0 + ko;
    v2f a  = *(const v2f*)&ldsH[m * 772 + kk];
    v2f bb = *(const v2f*)&Wrow[kk];
    acc = WMMA_F32(a, bb, acc);
  }

  const float bv  = b_lin[fcol + m];
  const int   i   = kS - 1 - s;  // reversed sequence index
  const int   rhi = (lane >> 4) * 8;
#pragma unroll
  for (int r = 0; r < 8; ++r) {
    int b = rowBase + rhi + r;
    float val = (i < T[b]) ? (acc[r] + bv) : -1.0f;  // PADDING_VALUE
    out[((size_t)b * kS + i) * kF + fcol + m] = val;
  }
}

// ---------------------------------------------------------------------------
extern "C" void kernel_launch(void* const* d_in, const int* in_sizes, int n_in,
                              void* d_out, int out_size, void* d_ws, size_t ws_size,
                              hipStream_t stream) {
  const float* H_o   = (const float*)d_in[0];
  const float* H_n   = (const float*)d_in[1];
  const int*   T     = (const int*)d_in[2];
  const float* W_ih  = (const float*)d_in[3];
  const float* W_hh  = (const float*)d_in[4];
  const float* b_ih  = (const float*)d_in[5];
  const float* b_hh  = (const float*)d_in[6];
  const float* W_lin = (const float*)d_in[7];
  const float* b_lin = (const float*)d_in[8];
  float* out = (float*)d_out;

  // workspace layout (floats): G[128*4608] | hBuf[128*768] | x0[128*768]
  float* Gmat = (float*)d_ws;
  float* hBuf = Gmat + (size_t)kB * kGW;
  float* x0   = hBuf + (size_t)kB * kH;

  init_x0<<<(kB * kH + 255) / 256, 256, 0, stream>>>(H_o, x0);

  for (int s = 0; s < kS; ++s) {
    const float* xs = (s == 0) ? x0  : hBuf;   // after step 0, x == h == h_prev
    const float* hs = (s == 0) ? H_n : hBuf;   // H_n[0] is contiguous (128,768)
    gru_gates<<<288, 128, kGatesLdsBytes, stream>>>(xs, hs, W_ih, W_hh, b_ih, b_hh, Gmat);
    gru_combine_project<<<8, 256, 0, stream>>>(Gmat, hs, hBuf, W_lin, b_lin, T, out, s);
  }
}